// DetectionLoss_51616916963357
// MI455X (gfx1250) — compile-verified
//
#include <hip/hip_runtime.h>
#include <math.h>

// Problem constants (match reference)
constexpr int   B_  = 32;
constexpr int   N_  = 25200;
constexpr int   M_  = 200;
constexpr int   C_  = 80;
constexpr float EPSF = 1e-7f;

constexpr float LAMBDA_COORD = 5.0f;
constexpr float LAMBDA_OBJ   = 1.0f;
constexpr float LAMBDA_NOOBJ = 0.5f;
constexpr float LAMBDA_CLS   = 1.0f;

typedef __attribute__((ext_vector_type(2))) float v2f;
typedef __attribute__((ext_vector_type(8))) float v8f;

// Numerically stable softplus: log(1 + e^t) = max(t,0) + log1p(e^{-|t|})
__device__ __forceinline__ float softplusf(float t) {
    return fmaxf(t, 0.0f) + log1pf(expf(-fabsf(t)));
}

// Wave-level sum reduction via one V_WMMA_F32_16X16X4_F32.
// A: lane L<16 supplies A[L][0]=v (K1=0); lane L>=16 supplies A[L-16][2]=v (K3=0).
// B: all-ones 4x16 (layout independent). D row m (replicated over N) = v[m]+v[m+16].
// Lane in 0..15 sums D VGPRs 0..7 -> rows 0..7; lane in 16..31 -> rows 8..15.
// So lane 0 + lane 16 atomics deliver the full 32-lane sum.
// Requires EXEC == all ones: call only from uniform control flow with full blocks.
__device__ __forceinline__ void waveReduceAtomicAdd(float v, float* dst) {
    v2f a; a.x = v;    a.y = 0.0f;
    v2f b; b.x = 1.0f; b.y = 1.0f;
    v8f c = {0.f, 0.f, 0.f, 0.f, 0.f, 0.f, 0.f, 0.f};
    c = __builtin_amdgcn_wmma_f32_16x16x4_f32(
            /*neg_a=*/false, a, /*neg_b=*/false, b,
            /*c_mod=*/(short)0, c, /*reuse_a=*/false, /*reuse_b=*/false);
    float s = ((c[0] + c[1]) + (c[2] + c[3])) + ((c[4] + c[5]) + (c[6] + c[7]));
    unsigned lane = threadIdx.x & 31u;
    if (lane == 0u || lane == 16u) {
        atomicAdd(dst, s);
    }
}

// ws[0] = loss_obj (pre-scaled sum), ws[1] = loss_bbox, ws[2] = loss_cls
__global__ void init_ws_kernel(float* __restrict__ ws) {
    if (threadIdx.x < 4) ws[threadIdx.x] = 0.0f;
}

// ---------------- Objectness: streams all of pred_obj (3.2 MB) ----------------
__global__ void obj_loss_kernel(const float* __restrict__ pred_obj,
                                float* __restrict__ ws) {
    const float sp = LAMBDA_OBJ   / (float)(B_ * M_);          // pos scale
    const float sn = LAMBDA_NOOBJ / (float)(B_ * (N_ - M_));   // neg scale
    const int total  = B_ * N_;
    const int stride = gridDim.x * blockDim.x;
    float acc = 0.0f;
    for (int i = blockIdx.x * blockDim.x + threadIdx.x; i < total; i += stride) {
        // gfx1250 global_prefetch_b8: speculative, safe past end (dropped on fault)
        __builtin_prefetch(pred_obj + i + 4 * stride, 0, 0);
        float x = pred_obj[i];
        int   n = i % N_;
        acc += (n < M_) ? (sp * softplusf(-x)) : (sn * softplusf(x));
    }
    waveReduceAtomicAdd(acc, ws + 0);
}

// ---------------- GIoU bbox loss: 32x200 elements ----------------
__global__ void bbox_loss_kernel(const float* __restrict__ pred_bbox,
                                 const float* __restrict__ gt_boxes,
                                 float* __restrict__ ws) {
    const float scale = LAMBDA_COORD / (float)(B_ * M_);
    const int total  = B_ * M_;
    const int stride = gridDim.x * blockDim.x;
    float acc = 0.0f;
    for (int i = blockIdx.x * blockDim.x + threadIdx.x; i < total; i += stride) {
        int b = i / M_;
        int m = i - b * M_;
        float4 p = *(const float4*)(pred_bbox + ((size_t)b * N_ + m) * 4);
        float4 t = *(const float4*)(gt_boxes + (size_t)i * 4);

        float px1 = p.x - p.z * 0.5f, px2 = p.x + p.z * 0.5f;
        float py1 = p.y - p.w * 0.5f, py2 = p.y + p.w * 0.5f;
        float tx1 = t.x - t.z * 0.5f, tx2 = t.x + t.z * 0.5f;
        float ty1 = t.y - t.w * 0.5f, ty2 = t.y + t.w * 0.5f;

        float iw    = fmaxf(fminf(px2, tx2) - fmaxf(px1, tx1), 0.0f);
        float ih    = fmaxf(fminf(py2, ty2) - fmaxf(py1, ty1), 0.0f);
        float inter = iw * ih;
        float uni   = (px2 - px1) * (py2 - py1) + (tx2 - tx1) * (ty2 - ty1) - inter;
        float iou   = inter / (uni + EPSF);

        float ew      = fmaxf(px2, tx2) - fminf(px1, tx1);
        float eh      = fmaxf(py2, ty2) - fminf(py1, ty1);
        float enclose = ew * eh;

        float giou_loss = 1.0f - (iou - (enclose - uni) / (enclose + EPSF));
        acc += scale * giou_loss;
    }
    waveReduceAtomicAdd(acc, ws + 1);
}

// ---------------- Class NLL: one wave per (b,m) row of 80 logits ----------------
__global__ void cls_loss_kernel(const float* __restrict__ pred_cls,
                                const int* __restrict__ gt_labels,
                                float* __restrict__ ws) {
    const float scale = LAMBDA_CLS / (float)(B_ * M_);
    const int lane    = threadIdx.x & 31;
    const int wid     = (blockIdx.x * blockDim.x + threadIdx.x) >> 5;
    const int nwaves  = (gridDim.x * blockDim.x) >> 5;
    const int rows    = B_ * M_;

    float acc = 0.0f;  // accumulated on lane 0 only
    for (int row = wid; row < rows; row += nwaves) {
        int b = row / M_;
        int m = row - b * M_;
        const float* r = pred_cls + ((size_t)b * N_ + m) * (size_t)C_;
        int label = gt_labels[row];

        // lane covers classes lane, lane+32, (lane+64 if < 80)
        float x0 = r[lane];
        float x1 = r[lane + 32];
        float x2 = (lane < C_ - 64) ? r[lane + 64] : -INFINITY;

        float mx = fmaxf(fmaxf(x0, x1), x2);
        #pragma unroll
        for (int off = 16; off > 0; off >>= 1)
            mx = fmaxf(mx, __shfl_xor(mx, off, 32));

        float se = expf(x0 - mx) + expf(x1 - mx)
                 + ((lane < C_ - 64) ? expf(x2 - mx) : 0.0f);
        float tl = 0.0f;
        if (lane == label)            tl = x0;
        else if (lane + 32 == label)  tl = x1;
        else if (lane + 64 == label)  tl = x2;

        #pragma unroll
        for (int off = 16; off > 0; off >>= 1) {
            se += __shfl_xor(se, off, 32);
            tl += __shfl_xor(tl, off, 32);
        }
        // all lanes now agree; accumulate once per wave
        if (lane == 0) acc += scale * ((mx + logf(se)) - tl);
    }
    if (lane == 0) atomicAdd(ws + 2, acc);
}

// ---------------- Finalize: out = [total, bbox, obj, cls] ----------------
__global__ void finalize_kernel(const float* __restrict__ ws,
                                float* __restrict__ out) {
    if (blockIdx.x == 0 && threadIdx.x == 0) {
        float lobj = ws[0], lbbox = ws[1], lcls = ws[2];
        out[0] = lbbox + lobj + lcls;
        out[1] = lbbox;
        out[2] = lobj;
        out[3] = lcls;
    }
}

extern "C" void kernel_launch(void* const* d_in, const int* in_sizes, int n_in,
                              void* d_out, int out_size, void* d_ws, size_t ws_size,
                              hipStream_t stream) {
    const float* pred_bbox = (const float*)d_in[0];  // (B, N, 4)
    const float* pred_obj  = (const float*)d_in[1];  // (B, N)
    const float* pred_cls  = (const float*)d_in[2];  // (B, N, 80)
    const float* gt_boxes  = (const float*)d_in[3];  // (B, M, 4)
    const int*   gt_labels = (const int*)d_in[4];    // (B, M)
    float* out = (float*)d_out;
    float* ws  = (float*)d_ws;

    init_ws_kernel<<<1, 32, 0, stream>>>(ws);

    // 806400 floats streamed; 512 blocks x 256 threads, ~6 elems/thread
    obj_loss_kernel<<<512, 256, 0, stream>>>(pred_obj, ws);

    // 6400 elements, exactly one per thread (full blocks keep EXEC all-ones)
    bbox_loss_kernel<<<25, 256, 0, stream>>>(pred_bbox, gt_boxes, ws);

    // 6400 rows, 8 waves/block -> one row per wave
    cls_loss_kernel<<<800, 256, 0, stream>>>(pred_cls, gt_labels, ws);

    finalize_kernel<<<1, 1, 0, stream>>>(ws, out);
}